// WholeBrainFastDMF_18717467476555
// MI455X (gfx1250) — compile-verified
//
#include <hip/hip_runtime.h>
#include <hip/hip_bf16.h>

typedef __attribute__((ext_vector_type(2))) float v2f;
typedef __attribute__((ext_vector_type(8))) float v8f;

#define N_ 360
#define T_ 7500
#define B_ 8
#define RATIO_ 10
#define HEMO_ 750

// Persistent whole-simulation kernel: one workgroup per batch.
// E/conn in LDS, sc resident in L2, conn via V_WMMA_F32_16X16X4_F32.
__launch_bounds__(736, 1)
__global__ void dmf_persistent_kernel(const float* __restrict__ state,
                                      const float* __restrict__ noise_in,
                                      const float* __restrict__ noise_out,
                                      const float* __restrict__ sc,
                                      float* __restrict__ out)
{
    __shared__ __align__(16) float Elds[N_];   // current E per region (this batch)
    __shared__ __align__(16) float Clds[N_];   // conn per region

    const int b    = blockIdx.x;        // batch
    const int tid  = threadIdx.x;       // 0..735 (23 waves)
    const int lane = tid & 31;
    const int wave = tid >> 5;          // 0..22 -> 16-row conn tile each
    const int hlf  = lane >> 4;         // 0: K=0,1  1: K=2,3  (A/B layout halves)
    const int l15  = lane & 15;         // M index within tile (A); N column (B/D)

    const float* scB = sc + (size_t)b * N_ * N_;

    // ---------------- per-region private state ----------------
    float e = 0.f, iv = 0.f, s = 0.f, f = 0.f, vv = 0.f, q = 0.f, Ji = 0.f;
    const float* np = nullptr;          // this region's noise stream
    if (tid < N_) {
        const int n = tid;
        e  = state[(n*6+0)*B_ + b];
        iv = state[(n*6+1)*B_ + b];
        s  = state[(n*6+2)*B_ + b];
        f  = state[(n*6+3)*B_ + b];
        vv = state[(n*6+4)*B_ + b];
        q  = state[(n*6+5)*B_ + b];
        // Ji[n,b] = 1 + 0.75*G*sum_m sc[b,n,m]
        float rs = 0.f;
        const float* row = scB + (size_t)n * N_;
        for (int m = 0; m < N_; ++m) rs += row[m];
        Ji = 1.0f + 1.5f * rs;
        Elds[tid] = e;
        // noise_in layout (N,T,2,B): idx = (n*T+t)*16 + c*8 + b
        np = noise_in + (size_t)n * T_ * 16 + b;
    }
    __syncthreads();

    // A-matrix row pointer for this lane (clamped for tail wave; extra rows discarded)
    int rowA = wave * 16 + l15;
    if (rowA > N_ - 1) rowA = N_ - 1;
    const float* arow = scB + (size_t)rowA * N_ + hlf * 2;

    // D-column-0 extraction: lane 0 holds M=0..7, lane 16 holds M=8..15.
    // base is a multiple of 8 and either fully in range or fully out.
    const int  dbase   = wave * 16 + hlf * 8;
    const bool dwriter = (l15 == 0) && (dbase < N_);

    int t = 0;
    for (int h = 0; h < HEMO_; ++h) {
        for (int r = 0; r < RATIO_; ++r, ++t) {
            // Issue this step's noise loads EARLY: their latency hides
            // under the 90-iteration WMMA matvec below.
            float En = 0.f, In = 0.f;
            if (tid < N_) {
                En = np[(size_t)t * 16];
                In = np[(size_t)t * 16 + 8];
                if ((t & 7) == 0)
                    __builtin_prefetch(np + (size_t)(t + 8) * 16, 0, 1);
            }

            // ---- conn = sc_b @ E  (fp32 WMMA 16x16x4, accumulate over K=360) ----
            v8f acc = {};
            #pragma unroll 6
            for (int m0 = 0; m0 < N_; m0 += 4) {
                v2f a = *(const v2f*)(arow + m0);          // sc[row][m0+2*hlf +0/1]
                float e0 = Elds[m0 + hlf * 2];
                float e1 = Elds[m0 + hlf * 2 + 1];
                v2f bm;                                     // B column 0 only
                bm.x = (l15 == 0) ? e0 : 0.0f;
                bm.y = (l15 == 0) ? e1 : 0.0f;
                acc = __builtin_amdgcn_wmma_f32_16x16x4_f32(
                        false, a, false, bm, (short)0, acc, false, false);
            }
            // Two ds_store_b128 instead of 8 predicated scalar stores.
            if (dwriter) {
                float4* dst = (float4*)&Clds[dbase];
                dst[0] = make_float4(acc[0], acc[1], acc[2], acc[3]);
                dst[1] = make_float4(acc[4], acc[5], acc[6], acc[7]);
            }
            __syncthreads();

            // ---- elementwise neural update ----
            if (tid < N_) {
                float cn  = Clds[tid];
                float I_E = fmaxf(0.f, 0.382f + 1.4f*0.15f*e + 2.0f*0.15f*cn - Ji*iv);
                float I_I = fmaxf(0.f, 0.7f*0.382f + 0.15f*e - iv);
                float xE  = 310.f*I_E - 125.f;
                float rE  = xE / (1.f - __expf(-0.16f*xE) + 1e-8f);
                float xI  = 615.f*I_I - 177.f;
                float rI  = xI / (1.f - __expf(-0.087f*xI) + 1e-8f);
                float dE  = -e*(1.f/100.f) + (1.f - e)*(0.641f/1000.f)*rE;
                float dI  = -iv*(1.f/10.f) + (1.f/1000.f)*rI;
                e  = fmaxf(0.f, e  + 0.1f*dE + 0.005f*En*0.31622776601683794f);
                iv = fmaxf(0.f, iv + 0.1f*dI + 0.005f*In*0.31622776601683794f);
                Elds[tid] = e;
            }
            __syncthreads();
        }
        // ---- hemodynamic update (every RATIO neural steps) ----
        if (tid < N_) {
            float v_ia = __powf(vv, 3.125f);                 // v^(1/0.32)
            float ds_  = e - s*(1.f/0.65f) - (f - 1.f)*(1.f/0.41f);
            float df_  = s;
            float dv_  = (f - v_ia)*(1.f/0.98f);
            float dq_  = (f*(1.f/0.34f)*(1.f - __powf(0.66f, 1.f/f))
                          - q*v_ia/vv)*(1.f/0.98f);
            s  += 1e-3f*ds_;
            f  += 1e-3f*df_;
            vv += 1e-3f*dv_;
            q  += 1e-3f*dq_;
        }
    }

    // ---- outputs: next_state (N,6,B) then bold (N,B), concatenated flat ----
    if (tid < N_) {
        const int n = tid;
        out[(n*6+0)*B_ + b] = e;
        out[(n*6+1)*B_ + b] = iv;
        out[(n*6+2)*B_ + b] = s;
        out[(n*6+3)*B_ + b] = f;
        out[(n*6+4)*B_ + b] = vv;
        out[(n*6+5)*B_ + b] = q;
        float bold = 100.f*0.2f*(2.38f*(1.f - q) + 2.0f*(1.f - q/vv)
                                 + 0.48f*(1.f - vv))
                   + noise_out[n*B_ + b];
        out[N_*6*B_ + n*B_ + b] = bold;
    }
}

extern "C" void kernel_launch(void* const* d_in, const int* in_sizes, int n_in,
                              void* d_out, int out_size, void* d_ws, size_t ws_size,
                              hipStream_t stream) {
    const float* state     = (const float*)d_in[0];
    /* d_in[1] = delays — unused by the reference math */
    const float* noise_in  = (const float*)d_in[2];
    const float* noise_out = (const float*)d_in[3];
    const float* sc        = (const float*)d_in[4];
    float* out = (float*)d_out;
    (void)in_sizes; (void)n_in; (void)d_ws; (void)ws_size; (void)out_size;

    dmf_persistent_kernel<<<dim3(B_), dim3(736), 0, stream>>>(
        state, noise_in, noise_out, sc, out);
}